// ScatterGather_68736656605663
// MI455X (gfx1250) — compile-verified
//
#include <hip/hip_runtime.h>
#include <hip/hip_bf16.h>

// ScatterGather round trip: scatter by stable-argsort(route) followed by the
// inverse gather is the identity permutation on values -> out == x exactly.
// Roofline: 0 FLOPs, 128 MiB read + 128 MiB write = 256 MiB HBM traffic,
// ~11.5 us at 23.3 TB/s. Pure data movement -> use the CDNA5 async LDS
// data-mover path (ASYNCcnt-tracked), software-pipelined:
//   - 4 async loads issued back-to-back per thread (4 in flight per wave)
//   - drain with decreasing s_wait_asynccnt thresholds; async loads complete
//     in-order among loads, so "asynccnt <= (#younger loads)" proves the
//     oldest load has landed in LDS.
//   - each (thread, k) pair owns a dedicated LDS slot -> zero reuse hazards.
// The instruction IOFFSET is added to BOTH the LDS and global addresses, so
// with LDS slot(t,k) = t*16 + k*4096 and global chunks 4096 B apart, a single
// LDS register and a single 64-bit global address serve all four transfers
// via offset:0/4096/8192/12288.

#define A_LOAD(LDS, GA, OFF)                                                  \
  asm volatile("global_load_async_to_lds_b128 %0, %1, off offset:" #OFF      \
               :: "v"(LDS), "v"(GA) : "memory")
#define A_STORE(GB, LDS, OFF)                                                 \
  asm volatile("global_store_async_from_lds_b128 %0, %1, off offset:" #OFF   \
               :: "v"(GB), "v"(LDS) : "memory")
#define A_WAIT(N) asm volatile("s_wait_asynccnt " #N ::: "memory")

__global__ void ScatterGather_68736656605663_copy_async(
    const float* __restrict__ x, float* __restrict__ out, long long n) {
  extern __shared__ char smem[];  // 256 threads * 4 slots * 16 B = 16 KiB
  const long long n4 = n >> 2;    // float4 vector count

  // Block b owns vectors [b*1024, b*1024 + 1024); thread t handles vector
  // indices vbase + k*256 for k = 0..3 (each k-chunk is 4 KiB contiguous).
  const long long vbase = (long long)blockIdx.x * 1024 + threadIdx.x;
  const unsigned  lds   = (unsigned)threadIdx.x * 16u;  // slot(t,0)

  const float4* __restrict__ src4 = (const float4*)x;
  float4* __restrict__ dst4       = (float4*)out;

  unsigned long long ga = (unsigned long long)(const void*)(src4 + vbase);
  unsigned long long gb = (unsigned long long)(void*)(dst4 + vbase);

  if ((long long)blockIdx.x * 1024 + 1023 < n4) {
    // Fast path: all 1024 vectors of this block are in range.
    A_LOAD(lds, ga, 0);
    A_LOAD(lds, ga, 4096);
    A_LOAD(lds, ga, 8192);
    A_LOAD(lds, ga, 12288);
    // asynccnt <= 3: if load0 were outstanding, loads 1..3 would be too
    // (in-order among loads) => >= 4 outstanding. So load0 has landed.
    A_WAIT(3);  A_STORE(gb, lds, 0);
    A_WAIT(2);  A_STORE(gb, lds, 4096);
    A_WAIT(1);  A_STORE(gb, lds, 8192);
    A_WAIT(0);  A_STORE(gb, lds, 12288);
  } else {
    // Tail block (unused for the reference shapes): conservative per-element
    // load -> wait 0 -> store.
    if (vbase          < n4) { A_LOAD(lds, ga, 0);     A_WAIT(0); A_STORE(gb, lds, 0);     }
    if (vbase + 256    < n4) { A_LOAD(lds, ga, 4096);  A_WAIT(0); A_STORE(gb, lds, 4096);  }
    if (vbase + 512    < n4) { A_LOAD(lds, ga, 8192);  A_WAIT(0); A_STORE(gb, lds, 8192);  }
    if (vbase + 768    < n4) { A_LOAD(lds, ga, 12288); A_WAIT(0); A_STORE(gb, lds, 12288); }
    // Scalar tail for n % 4 != 0 (empty here), handled by block 0 only.
    if (blockIdx.x == 0) {
      for (long long j = (n4 << 2) + threadIdx.x; j < n; j += blockDim.x) {
        out[j] = x[j];
      }
    }
  }

  // Outstanding async stores drain at s_endpgm (implicit wait-idle); keep an
  // explicit barrier against compiler sinking.
  A_WAIT(0);
}

extern "C" void kernel_launch(void* const* d_in, const int* in_sizes, int n_in,
                              void* d_out, int out_size, void* d_ws, size_t ws_size,
                              hipStream_t stream) {
  // Inputs: d_in[0] = x (float32, B*T*D); d_in[1] = route (int64, unused);
  //         d_in[2] = n_routes (scalar, unused). Output == x (see header).
  const float* x = (const float*)d_in[0];
  float*     out = (float*)d_out;
  const long long n  = (long long)in_sizes[0];  // 33,554,432 floats
  const long long n4 = n >> 2;                  // 8,388,608 float4 vectors

  const int threads = 256;                      // 8 wave32 waves
  long long blocks  = (n4 + 1023) / 1024;       // 1024 vectors per block
  if (blocks < 1) blocks = 1;
  const size_t lds_bytes = 16384;               // 4 slots * 16 B * 256 threads

  ScatterGather_68736656605663_copy_async<<<(int)blocks, threads, lds_bytes, stream>>>(
      x, out, n);
}